// AlephPracticalEngine_86466281603507
// MI455X (gfx1250) — compile-verified
//
#include <hip/hip_runtime.h>
#include <stdint.h>

#ifndef __has_builtin
#define __has_builtin(x) 0
#endif

#define DIM     4096
#define BATCH   2048
#define KENG    40
#define ROWS    8            // batch rows per workgroup
#define THREADS 256
#define NWAVE   (THREADS/32)

typedef float        v2f __attribute__((ext_vector_type(2)));
typedef float        v8f __attribute__((ext_vector_type(8)));
typedef unsigned int v4u __attribute__((ext_vector_type(4)));
typedef int          v8i __attribute__((ext_vector_type(8)));
typedef int          v4i __attribute__((ext_vector_type(4)));

#if __has_builtin(__builtin_amdgcn_wmma_f32_16x16x4_f32)
#define HAVE_WMMA_F32 1
#else
#define HAVE_WMMA_F32 0
#endif

#if __has_builtin(__builtin_amdgcn_tensor_load_to_lds) && __has_builtin(__builtin_amdgcn_s_wait_tensorcnt)
#define HAVE_TDM 1
#else
#define HAVE_TDM 0
#endif

// T16 = composition of the last four fhh butterfly stages (block sizes 16,8,4,2),
// derived analytically from the recursion  T_n = (T_{n/2} (+) T_{n/2}) * B_n,
// B_n = [[I, I], [(I+Roll1), -(I+Roll1)]].
__constant__ float T16[16][16] = {
  { 1, 1, 1, 1,  1, 1, 1, 1,  1, 1, 1, 1,  1, 1, 1, 1},
  { 2,-2, 2,-2,  2,-2, 2,-2,  2,-2, 2,-2,  2,-2, 2,-2},
  { 2, 2,-2,-2,  2, 2,-2,-2,  2, 2,-2,-2,  2, 2,-2,-2},
  { 0, 0, 0, 0,  0, 0, 0, 0,  0, 0, 0, 0,  0, 0, 0, 0},
  { 2, 2, 2, 2, -2,-2,-2,-2,  2, 2, 2, 2, -2,-2,-2,-2},
  { 0, 0, 0, 0,  0, 0, 0, 0,  0, 0, 0, 0,  0, 0, 0, 0},
  { 4, 0,-4, 0, -4, 0, 4, 0,  4, 0,-4, 0, -4, 0, 4, 0},
  { 0, 0, 0, 0,  0, 0, 0, 0,  0, 0, 0, 0,  0, 0, 0, 0},
  { 2, 2, 2, 2,  2, 2, 2, 2, -2,-2,-2,-2, -2,-2,-2,-2},
  { 0, 0, 0, 0,  0, 0, 0, 0,  0, 0, 0, 0,  0, 0, 0, 0},
  { 4, 0,-4, 0,  4, 0,-4, 0, -4, 0, 4, 0, -4, 0, 4, 0},
  { 0, 0, 0, 0,  0, 0, 0, 0,  0, 0, 0, 0,  0, 0, 0, 0},
  { 4, 4, 4, 0, -4,-4,-4, 0, -4,-4,-4, 0,  4, 4, 4, 0},
  { 0, 0, 0, 0,  0, 0, 0, 0,  0, 0, 0, 0,  0, 0, 0, 0},
  { 4,-4,-4,-4, -4, 4, 4, 4, -4, 4, 4, 4,  4,-4,-4,-4},
  { 0, 0, 0, 0,  0, 0, 0, 0,  0, 0, 0, 0,  0, 0, 0, 0},
};

// Eight butterfly stages (half = 2048 .. 16), ping-pong a<->b, result ends in 'a'.
// Each thread owns one contiguous run of 8 pairs per row; the roll(diff,1)
// neighbour is recomputed from source (one extra pair read) instead of
// communicated, so a single barrier per stage suffices.
__device__ __forceinline__ void fhh_stages8(float* __restrict__ a,
                                            float* __restrict__ b, int tid)
{
  float* src = a;
  float* dst = b;
#pragma unroll
  for (int half = DIM / 2; half >= 16; half >>= 1) {
    const int p0   = tid << 3;                 // first pair index (of 2048)
    const int i0   = p0 & (half - 1);          // offset inside half-block
    const int base = (p0 & ~(half - 1)) << 1;  // block start
    const int im1  = (i0 - 1) & (half - 1);    // wrapped neighbour
#pragma unroll
    for (int r = 0; r < ROWS; r++) {
      const float* sp = src + r * DIM + base;
      float*       dp = dst + r * DIM + base;
      float prevd = sp[im1] - sp[im1 + half];
#pragma unroll
      for (int j = 0; j < 8; j++) {
        const float l  = sp[i0 + j];
        const float rr = sp[i0 + j + half];
        dp[i0 + j] = l + rr;
        const float df = l - rr;
        dp[i0 + j + half] = df + prevd;
        prevd = df;
      }
    }
    __syncthreads();
    float* t = src; src = dst; dst = t;
  }
}

// Last four stages as Y_tile = T16 * X_tile via 4 chained V_WMMA_F32_16X16X4_F32.
// X[i][c] = src[16c + i] over 16 chunks per tile, 16 tiles per row.
// Static trip count: loop is provably wave-uniform -> no exec masking around WMMA.
__device__ __forceinline__ void fhh_tail16(const float* __restrict__ src,
                                           float* __restrict__ dst, int tid)
{
  const int lane = tid & 31;
  const int wv   = tid >> 5;
  const int n    = lane & 15;   // column (chunk) index / A-matrix row m
  const int hi   = lane >> 4;   // lane half selects K sub-pair (ISA 7.12.2)

  v2f a[4];
#pragma unroll
  for (int k = 0; k < 4; k++) {           // A operand: T16 columns 4k..4k+3
    a[k].x = T16[n][4 * k + 2 * hi];
    a[k].y = T16[n][4 * k + 2 * hi + 1];
  }

#pragma unroll
  for (int it = 0; it < (ROWS * 16) / NWAVE; it++) {
    const int t    = it * NWAVE + wv;
    const int r    = t >> 4;
    const int tile = t & 15;
    const float* base = src + r * DIM + tile * 256;
    v8f acc = {0.f, 0.f, 0.f, 0.f, 0.f, 0.f, 0.f, 0.f};
#pragma unroll
    for (int k = 0; k < 4; k++) {
#if HAVE_WMMA_F32
      v2f bmat;                                   // B[K][N], K rows 4k..4k+3
      bmat.x = base[n * 16 + 4 * k + 2 * hi];
      bmat.y = base[n * 16 + 4 * k + 2 * hi + 1];
      acc = __builtin_amdgcn_wmma_f32_16x16x4_f32(
          false, a[k], false, bmat, (short)0, acc, false, false);
#else
#pragma unroll
      for (int v = 0; v < 8; v++) {
        const int j = v + 8 * hi;
        float s4 = 0.f;
#pragma unroll
        for (int kk = 0; kk < 4; kk++)
          s4 += T16[j][4 * k + kk] * base[n * 16 + 4 * k + kk];
        acc[v] += s4;
      }
#endif
    }
    // D layout: VGPR v holds rows v (lanes 0-15) and v+8 (lanes 16-31)
    float* dp = dst + r * DIM + tile * 256 + n * 16 + 8 * hi;
#pragma unroll
    for (int v = 0; v < 8; v++) dp[v] = acc[v];
  }
}

__global__ __launch_bounds__(THREADS, 1)
void aleph_fhh_engine(const float* __restrict__ x,
                      const float* __restrict__ sw,
                      const float* __restrict__ ev,
                      const int*   __restrict__ eidx,
                      float* __restrict__ out)
{
  __shared__ __align__(16) float bufA[ROWS * DIM];   // 128 KB
  __shared__ __align__(16) float bufB[ROWS * DIM];   // 128 KB

  const int tid  = threadIdx.x;
  const int wg   = blockIdx.x;
  const int lane = tid & 31;
  const int wv   = tid >> 5;

  const float* xrow = x + (size_t)wg * ROWS * DIM;

  // ---- phase 0: bulk-load 8 rows (128 KB tile) into LDS via the TDM ----
#if HAVE_TDM
  if (wv == 0) {
    const unsigned lds_off      = (unsigned)(uintptr_t)(void*)bufA;
    const unsigned long long ga = (unsigned long long)(uintptr_t)xrow;
    v4u g0;
    g0.x = 1u;                                           // count=1
    g0.y = lds_off;                                      // lds_addr [63:32]
    g0.z = (unsigned)ga;                                 // global_addr lo
    g0.w = (unsigned)((ga >> 32) & 0x01FFFFFFu) | (2u << 30);  // addr[56:32] | type=2
    v8i g1;
    g1[0] = (int)(2u << 16);                 // data_size = 4 bytes
    g1[1] = (int)((DIM & 0xFFFF) << 16);     // tensor_dim0 lo16 (bits 63:48)
    g1[2] = (int)((BATCH & 0xFFFF) << 16);   // dim0 hi16=0 | tensor_dim1 lo16
    g1[3] = (int)((DIM & 0xFFFF) << 16);     // dim1 hi16=0 | tile_dim0 = 4096
    g1[4] = ROWS;                            // tile_dim1 = 8 rows, tile_dim2 = 0
    g1[5] = DIM;                             // tensor_dim0_stride lo32
    g1[6] = 0;                               // stride0 hi16 | stride1 lo16
    g1[7] = 0;
    const v4i gz4 = {0, 0, 0, 0};            // 2D tensor: groups 2/3 unused
    const v8i gz8 = {0, 0, 0, 0, 0, 0, 0, 0};
    __builtin_amdgcn_tensor_load_to_lds(g0, g1, gz4, gz4, gz8, 0);
    __builtin_amdgcn_s_wait_tensorcnt(0);
  }
  __syncthreads();
#else
  {
    const float4* gx = (const float4*)xrow;
    float4*       la = (float4*)bufA;
    for (int i = tid; i < ROWS * DIM / 4; i += THREADS) la[i] = gx[i];
  }
  __syncthreads();
#endif

  // ---- FHH #1 : 8 LDS stages (result in bufA) + WMMA tail (wave -> bufB) ----
  fhh_stages8(bufA, bufB, tid);
  fhh_tail16(bufA, bufB, tid);
  __syncthreads();

  // ---- transpose wave into bufA as waveT[pos][r] for conflict-free gathers ----
  for (int p = tid; p < DIM; p += THREADS) {
#pragma unroll
    for (int r = 0; r < ROWS; r++) bufA[p * ROWS + r] = bufB[r * DIM + p];
  }
  __syncthreads();

  // ---- middle: processed[r][d] = sw[d]*wave[r][d] + sum_k ev[d][k]*wave[r][idx[d][k]]
  // lanes = (row r, k-phase kk): coefficient loads shared across the 8 row-lanes,
  // LDS gathers are 4 runs of 8 consecutive floats (near conflict-free).
  {
    const int r  = lane >> 2;
    const int kk = lane & 3;
    for (int it = 0; it < DIM / NWAVE; it++) {
      const int dd = it * NWAVE + wv;
      const float* evd = ev   + dd * KENG;
      const int*   ixd = eidx + dd * KENG;
      float acc = 0.f;
#pragma unroll
      for (int t = 0; t < KENG / 4; t++) {
        const int k = kk + 4 * t;
        acc += evd[k] * bufA[ixd[k] * ROWS + r];
      }
      acc += __shfl_xor(acc, 1, 32);        // reduce the 4 k-phases
      acc += __shfl_xor(acc, 2, 32);
      if (kk == 0)
        bufB[r * DIM + dd] = sw[dd] * bufA[dd * ROWS + r] + acc;
    }
  }
  __syncthreads();

  // ---- FHH #2 : stages (result in bufB) + WMMA tail (-> bufA) ----
  fhh_stages8(bufB, bufA, tid);
  fhh_tail16(bufB, bufA, tid);
  __syncthreads();

  // ---- scale by 1/DIM and store ----
  {
    const float sc = 1.0f / (float)DIM;
    float4*       go = (float4*)(out + (size_t)wg * ROWS * DIM);
    const float4* fa = (const float4*)bufA;
    for (int i = tid; i < ROWS * DIM / 4; i += THREADS) {
      float4 v = fa[i];
      v.x *= sc; v.y *= sc; v.z *= sc; v.w *= sc;
      go[i] = v;
    }
  }
}

extern "C" void kernel_launch(void* const* d_in, const int* in_sizes, int n_in,
                              void* d_out, int out_size, void* d_ws, size_t ws_size,
                              hipStream_t stream)
{
  (void)in_sizes; (void)n_in; (void)out_size; (void)d_ws; (void)ws_size;
  const float* x   = (const float*)d_in[0];
  const float* sw  = (const float*)d_in[1];
  const float* ev  = (const float*)d_in[2];
  const int*   idx = (const int*)d_in[3];
  float* out = (float*)d_out;

  dim3 grid(BATCH / ROWS);
  dim3 block(THREADS);
  hipLaunchKernelGGL(aleph_fhh_engine, grid, block, 0, stream, x, sw, ev, idx, out);
}